// GCNModel_43233140801624
// MI455X (gfx1250) — compile-verified
//
#include <hip/hip_runtime.h>

#define NN 100000      // nodes
#define NE 1600000     // edges
#define NG 2000        // graphs
#define INF 128        // in feats
#define HID 256
#define CLS 128

typedef float v2f __attribute__((ext_vector_type(2)));
typedef float v8f __attribute__((ext_vector_type(8)));

// ---------------- integer degrees ----------------
__global__ void deg_kernel(const int* __restrict__ src, const int* __restrict__ dst,
                           int* __restrict__ odeg, int* __restrict__ ideg) {
  int i = blockIdx.x * blockDim.x + threadIdx.x;
  if (i < NE) {
    atomicAdd(&odeg[src[i]], 1);
    atomicAdd(&ideg[dst[i]], 1);
  }
}

// norm = rsqrt(max(deg,1)) for both deg arrays (contiguous int -> contiguous float)
__global__ void norm_kernel(const int* __restrict__ deg, float* __restrict__ nrm, int n) {
  int i = blockIdx.x * blockDim.x + threadIdx.x;
  if (i < n) nrm[i] = rsqrtf(fmaxf((float)deg[i], 1.0f));
}

// ---------------- exclusive scan of in-degree -> row_ptr, cursor ----------------
// Single workgroup of 1024 threads; ~98 chunk iterations over 100K entries.
__global__ __launch_bounds__(1024) void scan_kernel(const int* __restrict__ deg,
                                                    int* __restrict__ row_ptr,
                                                    int* __restrict__ cursor) {
  __shared__ int lds[1024];
  __shared__ int carry;
  int tid = threadIdx.x;
  if (tid == 0) carry = 0;
  __syncthreads();
  for (int base = 0; base < NN; base += 1024) {
    int c = carry;                       // uniform read
    int i = base + tid;
    int v = (i < NN) ? deg[i] : 0;
    lds[tid] = v;
    __syncthreads();
    for (int off = 1; off < 1024; off <<= 1) {
      int t = (tid >= off) ? lds[tid - off] : 0;
      __syncthreads();
      lds[tid] += t;
      __syncthreads();
    }
    int incl = lds[tid];
    if (i < NN) {
      int rp = c + incl - v;             // exclusive
      row_ptr[i] = rp;
      cursor[i] = rp;
    }
    __syncthreads();
    if (tid == 1023) carry = c + incl;
    __syncthreads();
  }
  if (tid == 0) row_ptr[NN] = carry;     // == NE
}

// ---------------- bucket fill: group edges by destination ----------------
__global__ void fill_kernel(const int* __restrict__ src, const int* __restrict__ dst,
                            int* __restrict__ cursor, int* __restrict__ col_src) {
  int i = blockIdx.x * blockDim.x + threadIdx.x;
  if (i < NE) {
    int pos = atomicAdd(&cursor[dst[i]], 1);
    col_src[pos] = src[i];
  }
}

// ---------------- fp32 WMMA GEMM + row scale ----------------
// C[r, :] = (A[r, :] @ B) * norm_src[r]
// One wave computes a 16x64 tile: 4 accumulator chains of V_WMMA_F32_16X16X4_F32.
// Layouts per CDNA5 ISA 7.12.2:
//   A 16x4 : lanes 0-15 -> M=lane, v0=K0,v1=K1 ; lanes 16-31 -> v0=K2,v1=K3
//   B 4x16 : lanes 0-15 -> N=lane, v0=row K0, v1=row K1 ; lanes 16-31 -> rows K2,K3
//   C 16x16: vgpr i -> row i (lanes 0-15) / row 8+i (lanes 16-31), col = lane%16
__global__ void gemm_wmma(const float* __restrict__ A, const float* __restrict__ B,
                          const float* __restrict__ norm_src, float* __restrict__ C,
                          int K) {
  int wave = (blockIdx.x * blockDim.x + threadIdx.x) >> 5;
  int lane = threadIdx.x & 31;
  int rt = wave >> 2;                       // 6250 row tiles
  int cg = wave & 3;                        // 4 col groups of 64
  if (rt * 16 >= NN) return;                // wave-uniform; EXEC stays all-ones
  int r0 = rt * 16, c0 = cg * 64;
  int half = lane >> 4, lm = lane & 15;
  const float* __restrict__ arow = A + (size_t)(r0 + lm) * K;
  const float* __restrict__ bcol = B + c0 + lm;
  v8f acc[4] = {{}, {}, {}, {}};
#pragma unroll 2
  for (int k = 0; k < K; k += 4) {
    v2f a;
    a.x = arow[k + 2 * half];
    a.y = arow[k + 2 * half + 1];
    const float* b0 = bcol + (size_t)(k + 2 * half) * HID;
#pragma unroll
    for (int t = 0; t < 4; ++t) {
      v2f b;
      b.x = b0[16 * t];
      b.y = b0[16 * t + HID];
      acc[t] = __builtin_amdgcn_wmma_f32_16x16x4_f32(false, a, false, b, (short)0,
                                                     acc[t], false, false);
    }
  }
  float ns[8];
#pragma unroll
  for (int i = 0; i < 8; ++i) ns[i] = norm_src[r0 + 8 * half + i];
#pragma unroll
  for (int t = 0; t < 4; ++t)
#pragma unroll
    for (int i = 0; i < 8; ++i)
      C[(size_t)(r0 + 8 * half + i) * HID + c0 + 16 * t + lm] = acc[t][i] * ns[i];
}

// ---------------- CSR gather: H[n] = relu(norm_dst[n] * sum_{s in N(n)} HWs[s] + b) ----
// One 256-thread block per node; coalesced 1KB row reads, no atomics.
// Edge loop unrolled 4-way: 4 independent index loads + 4 row loads in flight
// per iteration (4x memory-level parallelism in the dominant, L2-bound kernel).
__global__ void gather_kernel(const float* __restrict__ HWs, const int* __restrict__ row_ptr,
                              const int* __restrict__ col_src,
                              const float* __restrict__ norm_dst,
                              const float* __restrict__ bias, float* __restrict__ H) {
  int n = blockIdx.x;
  int f = threadIdx.x;
  int j = row_ptr[n], end = row_ptr[n + 1];
  float a0 = 0.0f, a1 = 0.0f, a2 = 0.0f, a3 = 0.0f;
  for (; j + 3 < end; j += 4) {
    int s0 = col_src[j + 0];
    int s1 = col_src[j + 1];
    int s2 = col_src[j + 2];
    int s3 = col_src[j + 3];
    a0 += HWs[(size_t)s0 * HID + f];
    a1 += HWs[(size_t)s1 * HID + f];
    a2 += HWs[(size_t)s2 * HID + f];
    a3 += HWs[(size_t)s3 * HID + f];
  }
  for (; j < end; ++j) a0 += HWs[(size_t)col_src[j] * HID + f];
  float acc = (a0 + a1) + (a2 + a3);
  float v = acc * norm_dst[n] + bias[f];
  H[(size_t)n * HID + f] = fmaxf(v, 0.0f);
}

// ---------------- pooling: per-graph max / sum / count ----------------
__global__ void pool_kernel(const float* __restrict__ H, const int* __restrict__ gid,
                            float* __restrict__ gmax, float* __restrict__ gsum,
                            float* __restrict__ counts) {
  int idx = blockIdx.x * blockDim.x + threadIdx.x;
  if (idx >= NN * 64) return;
  int n = idx >> 6;
  int c = (idx & 63) << 2;
  int g = gid[n];
  const float4 v = *(const float4*)(H + (size_t)n * HID + c);
  float* sp = gsum + (size_t)g * HID + c;
  atomicAdd(sp + 0, v.x); atomicAdd(sp + 1, v.y);
  atomicAdd(sp + 2, v.z); atomicAdd(sp + 3, v.w);
  // H >= 0 after relu, so integer-bit compare == float compare; gmax init to 0
  int* mp = (int*)(gmax + (size_t)g * HID + c);
  atomicMax(mp + 0, __float_as_int(v.x));
  atomicMax(mp + 1, __float_as_int(v.y));
  atomicMax(mp + 2, __float_as_int(v.z));
  atomicMax(mp + 3, __float_as_int(v.w));
  if (c == 0) atomicAdd(&counts[g], 1.0f);
}

// ---------------- g_rep = w0*max + w1*avg (w = softmax(pool_w)) ----------------
__global__ void combine_kernel(const float* __restrict__ pool_w,
                               const float* __restrict__ gmax,
                               const float* __restrict__ gsum,
                               const float* __restrict__ counts,
                               float* __restrict__ grep) {
  int idx = blockIdx.x * blockDim.x + threadIdx.x;
  if (idx >= NG * HID) return;
  int g = idx >> 8;
  float p0 = pool_w[0], p1 = pool_w[1];
  float m = fmaxf(p0, p1);
  float e0 = __expf(p0 - m), e1 = __expf(p1 - m);
  float inv = 1.0f / (e0 + e1);
  float cnt = fmaxf(counts[g], 1.0f);
  grep[idx] = (e0 * inv) * gmax[idx] + (e1 * inv) * (gsum[idx] / cnt);
}

// ---------------- classifier ----------------
__global__ void fc1_kernel(const float* __restrict__ grep, const float* __restrict__ W,
                           const float* __restrict__ b, float* __restrict__ X) {
  int idx = blockIdx.x * blockDim.x + threadIdx.x;
  if (idx >= NG * CLS) return;
  int g = idx >> 7, j = idx & (CLS - 1);
  const float* row = grep + (size_t)g * HID;
  float s = b[j];
  for (int f = 0; f < HID; ++f) s += row[f] * W[f * CLS + j];
  X[idx] = fmaxf(s, 0.0f);
}

__global__ void fc2_kernel(const float* __restrict__ X, const float* __restrict__ W,
                           const float* __restrict__ b, float* __restrict__ out) {
  int idx = blockIdx.x * blockDim.x + threadIdx.x;
  if (idx >= NG * 2) return;
  int g = idx >> 1, c = idx & 1;
  const float* row = X + (size_t)g * CLS;
  float s = b[c];
  for (int j = 0; j < CLS; ++j) s += row[j] * W[j * 2 + c];
  out[idx] = s;
}

extern "C" void kernel_launch(void* const* d_in, const int* in_sizes, int n_in,
                              void* d_out, int out_size, void* d_ws, size_t ws_size,
                              hipStream_t stream) {
  const float* features = (const float*)d_in[0];
  const int*   src      = (const int*)d_in[1];
  const int*   dst      = (const int*)d_in[2];
  const int*   gid      = (const int*)d_in[3];
  const float* W1 = (const float*)d_in[4];  const float* b1 = (const float*)d_in[5];
  const float* W2 = (const float*)d_in[6];  const float* b2 = (const float*)d_in[7];
  const float* W3 = (const float*)d_in[8];  const float* b3 = (const float*)d_in[9];
  const float* pool_w = (const float*)d_in[10];
  const float* fc1_w = (const float*)d_in[11]; const float* fc1_b = (const float*)d_in[12];
  const float* fc2_w = (const float*)d_in[13]; const float* fc2_b = (const float*)d_in[14];
  float* out = (float*)d_out;

  // ---- workspace layout: int region first (padded to 16B), then floats ----
  int* iws = (int*)d_ws;
  int* odeg    = iws;                       // NN
  int* ideg    = odeg + NN;                 // NN
  int* row_ptr = ideg + NN;                 // NN+4 (uses NN+1; padded for alignment)
  int* cursor  = row_ptr + NN + 4;          // NN
  int* col_src = cursor + NN;               // NE
  // total ints = 2,000,004 -> 8,000,016 bytes, multiple of 16
  float* fws = (float*)(col_src + NE);
  float* norm_src = fws;                    // NN  (rsqrt out-degree)
  float* norm_dst = norm_src + NN;          // NN  (rsqrt in-degree)
  float* bufA = norm_dst + NN;              // NN*HID  (scaled hW)   16B aligned
  float* bufB = bufA + (size_t)NN * HID;    // NN*HID  (layer output) 16B aligned
  float* gmax = bufB + (size_t)NN * HID;    // NG*HID
  float* gsum = gmax + (size_t)NG * HID;    // NG*HID
  float* counts = gsum + (size_t)NG * HID;  // NG
  float* grep = counts + NG;                // NG*HID
  float* x1 = grep + (size_t)NG * HID;      // NG*CLS

  const int BT = 256;
  // zero accumulators (every call; graph-capture safe)
  hipMemsetAsync(odeg, 0, (size_t)2 * NN * sizeof(int), stream);
  hipMemsetAsync(gmax, 0, ((size_t)2 * NG * HID + NG) * sizeof(float), stream);

  // ---- build norms + CSR (dst-grouped edge list) ----
  deg_kernel<<<(NE + BT - 1) / BT, BT, 0, stream>>>(src, dst, odeg, ideg);
  norm_kernel<<<(2 * NN + BT - 1) / BT, BT, 0, stream>>>(odeg, norm_src, 2 * NN);
  scan_kernel<<<1, 1024, 0, stream>>>(ideg, row_ptr, cursor);
  fill_kernel<<<(NE + BT - 1) / BT, BT, 0, stream>>>(src, dst, cursor, col_src);

  const int gemm_waves = (NN / 16) * (HID / 64);          // 25000 waves
  const int gemm_blocks = gemm_waves * 32 / BT;           // 3125

  // ---- layer 1 ----
  gemm_wmma<<<gemm_blocks, BT, 0, stream>>>(features, W1, norm_src, bufA, INF);
  gather_kernel<<<NN, HID, 0, stream>>>(bufA, row_ptr, col_src, norm_dst, b1, bufB);
  // ---- layer 2 ----
  gemm_wmma<<<gemm_blocks, BT, 0, stream>>>(bufB, W2, norm_src, bufA, HID);
  gather_kernel<<<NN, HID, 0, stream>>>(bufA, row_ptr, col_src, norm_dst, b2, bufB);
  // ---- layer 3 ----
  gemm_wmma<<<gemm_blocks, BT, 0, stream>>>(bufB, W3, norm_src, bufA, HID);
  gather_kernel<<<NN, HID, 0, stream>>>(bufA, row_ptr, col_src, norm_dst, b3, bufB);

  // ---- readout + classifier ----
  pool_kernel<<<(NN * 64 + BT - 1) / BT, BT, 0, stream>>>(bufB, gid, gmax, gsum, counts);
  combine_kernel<<<(NG * HID + BT - 1) / BT, BT, 0, stream>>>(pool_w, gmax, gsum, counts, grep);
  fc1_kernel<<<(NG * CLS + BT - 1) / BT, BT, 0, stream>>>(grep, fc1_w, fc1_b, x1);
  fc2_kernel<<<(NG * 2 + BT - 1) / BT, BT, 0, stream>>>(x1, fc2_w, fc2_b, out);
}